// PointEmbeddingandGroup_5549097746760
// MI455X (gfx1250) — compile-verified
//
#include <hip/hip_runtime.h>

#define B_   8
#define N_   8192
#define S_   1024
#define K_   32
#define DIN  6
#define C1   64
#define C2   64
#define C3   128
#define RAD2 0.04f

typedef __attribute__((ext_vector_type(2))) float v2f;
typedef __attribute__((ext_vector_type(8))) float v8f;

__device__ __forceinline__ v8f wmma4(v2f a, v2f b, v8f c) {
  // V_WMMA_F32_16X16X4_F32 : D(16x16,f32) = A(16x4,f32) x B(4x16,f32) + C
  return __builtin_amdgcn_wmma_f32_16x16x4_f32(false, a, false, b, (short)0, c,
                                               false, false);
}

// ---------------------------------------------------------------------------
// Kernel 1: per-point MLP embed  f(65536,6) -> embed(65536,128), WMMA f32.
// 2 waves / block, 1 tile of 16 points per wave. All weights staged once in
// LDS *transposed* ([cout][cin], pitch 68 -> conflict-free ds_load_b64
// fragments), f-tile staged+zero-padded in LDS, single in-place staging
// buffer per wave (layer-1 accs held in registers across the overwrite).
// ---------------------------------------------------------------------------
__global__ __launch_bounds__(64) void embed_kernel(
    const float* __restrict__ f,
    const float* __restrict__ W0, const float* __restrict__ b0,
    const float* __restrict__ g0, const float* __restrict__ t0,
    const float* __restrict__ W1, const float* __restrict__ b1,
    const float* __restrict__ g1, const float* __restrict__ t1,
    const float* __restrict__ W2, const float* __restrict__ b2,
    const float* __restrict__ g2, const float* __restrict__ t2,
    float* __restrict__ embed)
{
  __shared__ float sW0T[64][12];        //  3,072 B  W0^T, K padded 6->8
  __shared__ float sW1T[64][68];        // 17,408 B  W1^T
  __shared__ float sW2T[128][68];       // 34,816 B  W2^T
  __shared__ float sStage[2][16][68];   //  8,704 B  per-wave activation tile
  __shared__ float sF[2][16][12];       //  1,536 B  per-wave f tile (pad->8)
                                        // total = 65,536 B

  const int t    = threadIdx.x;
  const int lane = t & 31;
  const int wid  = t >> 5;
  const int m    = lane & 15;           // A/C row, B/C column id
  const int half = lane >> 4;           // K-half selector (ISA layout)

  // ---- cooperative transposed weight staging (once per block) ----
  for (int e = t; e < 64 * 8; e += 64) {
    const int k = e >> 6, c = e & 63;
    sW0T[c][k] = (k < DIN) ? W0[k * C1 + c] : 0.f;
  }
  for (int e = t; e < 64 * 64; e += 64) {
    const int k = e >> 6, c = e & 63;
    sW1T[c][k] = W1[e];
  }
  for (int e = t; e < 64 * 128; e += 64) {
    const int k = e >> 7, c = e & 127;
    sW2T[c][k] = W2[e];
  }
  if (lane < 16) { sF[wid][lane][6] = 0.f; sF[wid][lane][7] = 0.f; }
  __syncthreads();

  // ---- hoisted per-column scale/offset: (h@W + b)*g + beta = h@W*g + off ----
  float s0a[4], o0a[4], s1a[4], o1a[4], s2a[8], o2a[8];
  #pragma unroll
  for (int i = 0; i < 4; ++i) {
    const int c0 = i * 16 + m;
    float g = g0[c0]; s0a[i] = g; o0a[i] = b0[c0] * g + t0[c0];
    float h = g1[c0]; s1a[i] = h; o1a[i] = b1[c0] * h + t1[c0];
  }
  #pragma unroll
  for (int i = 0; i < 8; ++i) {
    const int c0 = i * 16 + m;
    float g = g2[c0]; s2a[i] = g; o2a[i] = b2[c0] * g + t2[c0];
  }

  float (*st)[68] = sStage[wid];
  float (*fb)[12] = sF[wid];

  const int nTiles = (B_ * N_) / 16;    // 4096
  const int wg     = blockIdx.x * 2 + wid;
  const int nW     = gridDim.x * 2;

  for (int tile = wg; tile < nTiles; tile += nW) {
    const int p0 = tile * 16;

    // stage the 16x6 f tile (96 contiguous floats, 3 per lane)
    #pragma unroll
    for (int j = 0; j < 3; ++j) {
      const int e = j * 32 + lane;
      fb[e / 6][e % 6] = f[p0 * DIN + e];
    }

    // ---- layer 0: sF(16x8) @ W0(8x64) -> st ----
    #pragma unroll
    for (int i = 0; i < 4; ++i) {
      v8f acc = {0.f,0.f,0.f,0.f,0.f,0.f,0.f,0.f};
      #pragma unroll
      for (int k0 = 0; k0 < 8; k0 += 4) {
        const int k = k0 + 2 * half;
        const v2f a  = *(const v2f*)&fb[m][k];
        const v2f bm = *(const v2f*)&sW0T[i * 16 + m][k];
        acc = wmma4(a, bm, acc);
      }
      #pragma unroll
      for (int r = 0; r < 8; ++r) {
        float v = acc[r] * s0a[i] + o0a[i];
        st[r + 8 * half][i * 16 + m] = v > 0.f ? v : 0.f;
      }
    }

    // ---- layer 1: st(16x64) @ W1(64x64); accs in regs, then overwrite st ----
    v8f acc1[4];
    #pragma unroll
    for (int i = 0; i < 4; ++i) {
      v8f acc = {0.f,0.f,0.f,0.f,0.f,0.f,0.f,0.f};
      #pragma unroll
      for (int k0 = 0; k0 < C1; k0 += 4) {
        const int k = k0 + 2 * half;
        const v2f a  = *(const v2f*)&st[m][k];
        const v2f bm = *(const v2f*)&sW1T[i * 16 + m][k];
        acc = wmma4(a, bm, acc);
      }
      acc1[i] = acc;
    }
    #pragma unroll
    for (int i = 0; i < 4; ++i) {
      #pragma unroll
      for (int r = 0; r < 8; ++r) {
        float v = acc1[i][r] * s1a[i] + o1a[i];
        st[r + 8 * half][i * 16 + m] = v > 0.f ? v : 0.f;
      }
    }

    // ---- layer 2: st(16x64) @ W2(64x128) -> embed ----
    #pragma unroll
    for (int i = 0; i < 8; ++i) {
      v8f acc = {0.f,0.f,0.f,0.f,0.f,0.f,0.f,0.f};
      #pragma unroll
      for (int k0 = 0; k0 < C2; k0 += 4) {
        const int k = k0 + 2 * half;
        const v2f a  = *(const v2f*)&st[m][k];
        const v2f bm = *(const v2f*)&sW2T[i * 16 + m][k];
        acc = wmma4(a, bm, acc);
      }
      #pragma unroll
      for (int r = 0; r < 8; ++r) {
        float v = acc[r] * s2a[i] + o2a[i];
        v = v > 0.f ? v : 0.f;
        embed[(size_t)(p0 + r + 8 * half) * C3 + i * 16 + m] = v;
      }
    }
  }
}

// ---------------------------------------------------------------------------
// Kernel 2: farthest point sampling. 1 block / batch, 1024 threads,
// all dists in registers, wave32 shuffle argmax + tiny LDS stage.
// Records farthest BEFORE update (matches the reference scan).
// ---------------------------------------------------------------------------
__global__ __launch_bounds__(1024) void fps_kernel(const float* __restrict__ x,
                                                   int*   __restrict__ fps_idx,
                                                   float* __restrict__ xs)
{
  const int b = blockIdx.x;
  const int t = threadIdx.x;
  const float* xb = x + (size_t)b * N_ * 3;

  __shared__ float s_c[3];
  __shared__ int   s_cur;
  __shared__ float s_wv[32];
  __shared__ int   s_wi[32];

  float px[8], py[8], pz[8], dst[8];
  #pragma unroll
  for (int j = 0; j < 8; ++j) {
    const int i = j * 1024 + t;
    px[j] = xb[i * 3 + 0];
    py[j] = xb[i * 3 + 1];
    pz[j] = xb[i * 3 + 2];
    dst[j] = 1e10f;
  }
  if (t == 0) s_cur = 0;
  __syncthreads();

  for (int it = 0; it < S_; ++it) {
    if (t == 0) {
      const int c = s_cur;
      fps_idx[b * S_ + it] = c;
      const float cx = xb[c * 3 + 0];
      const float cy = xb[c * 3 + 1];
      const float cz = xb[c * 3 + 2];
      s_c[0] = cx; s_c[1] = cy; s_c[2] = cz;
      const int o = (b * S_ + it) * 3;
      xs[o + 0] = cx; xs[o + 1] = cy; xs[o + 2] = cz;
    }
    __syncthreads();
    const float cx = s_c[0], cy = s_c[1], cz = s_c[2];

    float bv = -1.f;
    int   bi = 0;
    #pragma unroll
    for (int j = 0; j < 8; ++j) {
      const float dx = px[j] - cx, dy = py[j] - cy, dz = pz[j] - cz;
      float d = dx * dx + dy * dy + dz * dz;
      d = fminf(dst[j], d);
      dst[j] = d;
      const int i = j * 1024 + t;
      if (d > bv || (d == bv && i < bi)) { bv = d; bi = i; }
    }
    #pragma unroll
    for (int off = 16; off > 0; off >>= 1) {
      const float ov = __shfl_down(bv, off, 32);
      const int   oi = __shfl_down(bi, off, 32);
      if (ov > bv || (ov == bv && oi < bi)) { bv = ov; bi = oi; }
    }
    if ((t & 31) == 0) { s_wv[t >> 5] = bv; s_wi[t >> 5] = bi; }
    __syncthreads();
    if (t < 32) {
      bv = s_wv[t]; bi = s_wi[t];
      #pragma unroll
      for (int off = 16; off > 0; off >>= 1) {
        const float ov = __shfl_down(bv, off, 32);
        const int   oi = __shfl_down(bi, off, 32);
        if (ov > bv || (ov == bv && oi < bi)) { bv = ov; bi = oi; }
      }
      if (t == 0) s_cur = bi;
    }
    __syncthreads();
  }
}

// ---------------------------------------------------------------------------
// Kernel 3: ball query. 1 wave per center; ballot + prefix-popc append of the
// first K in-index-order points inside the radius; pad with first hit.
// ---------------------------------------------------------------------------
__global__ __launch_bounds__(256) void qball_kernel(const float* __restrict__ x,
                                                    const float* __restrict__ xs,
                                                    int* __restrict__ idx)
{
  const int lane = threadIdx.x & 31;
  const int wid  = threadIdx.x >> 5;
  const int c    = blockIdx.x * 8 + wid;        // 0 .. B*S-1
  const int b    = c / S_;
  const float* xb = x + (size_t)b * N_ * 3;

  const float cx = xs[c * 3 + 0];
  const float cy = xs[c * 3 + 1];
  const float cz = xs[c * 3 + 2];

  int found = 0;
  int firstIdx = -1;
  for (int n0 = 0; n0 < N_ && found < K_; n0 += 32) {
    const int i = n0 + lane;
    const float dx = xb[i * 3 + 0] - cx;
    const float dy = xb[i * 3 + 1] - cy;
    const float dz = xb[i * 3 + 2] - cz;
    const float d2 = dx * dx + dy * dy + dz * dz;
    const bool inside = (d2 <= RAD2);
    const unsigned mask = (unsigned)__ballot(inside);
    if (firstIdx < 0 && mask) firstIdx = n0 + (int)__builtin_ctz(mask);
    const int prefix = __popc(mask & ((1u << lane) - 1u));
    if (inside) {
      const int pos = found + prefix;
      if (pos < K_) idx[c * K_ + pos] = i;
    }
    found += __popc(mask);
  }
  if (firstIdx < 0) firstIdx = 0;   // cannot happen (center is a point), guard anyway
  if (found > K_) found = K_;
  for (int k = found + lane; k < K_; k += 32) idx[c * K_ + k] = firstIdx;
}

// ---------------------------------------------------------------------------
// Kernel 4: gathers. 1 block per center: xg(32x3), fg(32x128, float4 rows),
// fs(128). fg is the bandwidth floor (~134 MB written).
// ---------------------------------------------------------------------------
__global__ __launch_bounds__(128) void gather_kernel(const float* __restrict__ x,
                                                     const float* __restrict__ embed,
                                                     const int*   __restrict__ idx,
                                                     const int*   __restrict__ fps_idx,
                                                     float* __restrict__ xg,
                                                     float* __restrict__ fg,
                                                     float* __restrict__ fs)
{
  const int c = blockIdx.x;     // b*S + s
  const int b = c / S_;
  const int t = threadIdx.x;

  __shared__ int sIdx[K_];
  if (t < K_) sIdx[t] = idx[c * K_ + t];
  __syncthreads();

  if (t < K_ * 3) {
    const int k = t / 3, d = t % 3;
    xg[(size_t)c * K_ * 3 + t] = x[((size_t)b * N_ + sIdx[k]) * 3 + d];
  }

  const float4* e4  = (const float4*)embed;
  float4*       fg4 = (float4*)(fg + (size_t)c * K_ * C3);
  #pragma unroll
  for (int e = t; e < (K_ * C3) / 4; e += 128) {   // 1024 float4
    const int k   = e >> 5;                        // 32 float4 per 128-ch row
    const int ch4 = e & 31;
    fg4[e] = e4[((size_t)b * N_ + sIdx[k]) * 32 + ch4];
  }

  const int fsi = fps_idx[c];
  fs[(size_t)c * C3 + t] = embed[((size_t)b * N_ + fsi) * C3 + t];
}

// ---------------------------------------------------------------------------
extern "C" void kernel_launch(void* const* d_in, const int* in_sizes, int n_in,
                              void* d_out, int out_size, void* d_ws, size_t ws_size,
                              hipStream_t stream) {
  const float* x  = (const float*)d_in[0];
  const float* f  = (const float*)d_in[1];
  const float* W0 = (const float*)d_in[2];
  const float* b0 = (const float*)d_in[3];
  const float* g0 = (const float*)d_in[4];
  const float* t0 = (const float*)d_in[5];
  const float* W1 = (const float*)d_in[6];
  const float* b1 = (const float*)d_in[7];
  const float* g1 = (const float*)d_in[8];
  const float* t1 = (const float*)d_in[9];
  const float* W2 = (const float*)d_in[10];
  const float* b2 = (const float*)d_in[11];
  const float* g2 = (const float*)d_in[12];
  const float* t2 = (const float*)d_in[13];

  float* out = (float*)d_out;
  float* xg = out;                                 // 8*1024*32*3   = 786432
  float* fg = out + 786432;                        // 8*1024*32*128 = 33554432
  float* xs = out + 786432 + 33554432;             // 8*1024*3      = 24576
  float* fs = out + 786432 + 33554432 + 24576;     // 8*1024*128    = 1048576

  char* ws = (char*)d_ws;
  float* embed  = (float*)ws;                          // 33,554,432 B
  int*   fpsI   = (int*)(ws + 33554432);               //     32,768 B
  int*   ballI  = (int*)(ws + 33554432 + 32768);       //  1,048,576 B

  embed_kernel<<<512, 64, 0, stream>>>(f,
                                       W0, b0, g0, t0,
                                       W1, b1, g1, t1,
                                       W2, b2, g2, t2,
                                       embed);
  fps_kernel<<<B_, 1024, 0, stream>>>(x, fpsI, xs);
  qball_kernel<<<(B_ * S_) / 8, 256, 0, stream>>>(x, xs, ballI);
  gather_kernel<<<B_ * S_, 128, 0, stream>>>(x, embed, ballI, fpsI, xg, fg, fs);
}